// GNNModel_with_leaky_ReLU_29154238005713
// MI455X (gfx1250) — compile-verified
//
#include <hip/hip_runtime.h>
#include <hip/hip_bf16.h>

// ---------------------------------------------------------------------------
// Fused hetero-SAGE forward for MI455X (gfx1250, wave32).
//   Phase 1: edge scatter (gather x_src row, f32 atomics into per-dst sums +
//            counts)  -- bandwidth/atomic bound, ~2 GB total traffic.
//   Phase 2: per-node fused kernel: mean = sums/max(cnt,1) staged to LDS,
//            h = mean @ W_l.T + x_dst @ W_r.T + b_l  via V_WMMA_F32_16X16X4_F32
//            (two WMMAs per K-step, shared f32 accumulator), ReLU, then the
//            128->1 head reduced across the 8 waves through LDS, leaky-ReLU,
//            scalar store. h never touches global memory.
// ---------------------------------------------------------------------------

typedef __attribute__((ext_vector_type(2))) float v2f;
typedef __attribute__((ext_vector_type(8))) float v8f;

#define N_SITES 50000
#define N_WELLS 200000
#define D 128
#define N_EDGES 1000000

// ---------------------------------------------------------------------------
// Phase 1: scatter-mean accumulation. One wave (32 lanes) per edge; each lane
// moves 4 consecutive floats of the 128-wide feature row.
// ---------------------------------------------------------------------------
__global__ __launch_bounds__(256) void scatter_edges_kernel(
    const float* __restrict__ x_src,
    const int* __restrict__ src,
    const int* __restrict__ dst,
    float* __restrict__ sums,
    float* __restrict__ cnt,
    int n_edges)
{
    long long t = (long long)blockIdx.x * blockDim.x + threadIdx.x;
    int e    = (int)(t >> 5);
    int lane = (int)(t & 31);
    if (e >= n_edges) return;

    int s = src[e];
    int d = dst[e];

    const float4 v = ((const float4*)(x_src + (size_t)s * D))[lane];
    float* o = sums + (size_t)d * D + lane * 4;
    atomicAdd(o + 0, v.x);
    atomicAdd(o + 1, v.y);
    atomicAdd(o + 2, v.z);
    atomicAdd(o + 3, v.w);
    if (lane == 0) atomicAdd(&cnt[d], 1.0f);
}

// ---------------------------------------------------------------------------
// Phase 2: fused node update.
//   block = 256 threads = 8 waves; block handles 16 nodes x all 128 features.
//   wave w computes the 16x16 tile (nodes x features n0=16w..16w+15) with
//   V_WMMA_F32_16X16X4_F32, K-loop over 128 in steps of 4.
//
// Fragment layouts (wave32, per CDNA5 ISA):
//   A (16x4, M x K):  lane<16 : row=lane,    vgpr v = K = k0+v
//                     lane>=16: row=lane-16, vgpr v = K = k0+2+v
//   B (4x16, K x N):  mirror of A with N in place of M
//   C/D (16x16):      vgpr r, lane<16 -> (M=r,   N=lane)
//                              lane>=16 -> (M=r+8, N=lane-16)
// ---------------------------------------------------------------------------
__global__ __launch_bounds__(256) void node_update_wmma_kernel(
    const float* __restrict__ sums,     // [n_nodes, 128] neighbor sums
    const float* __restrict__ cnt,      // [n_nodes] neighbor counts (f32)
    const float* __restrict__ x_dst,    // [n_nodes, 128] root features
    const float* __restrict__ W_l,      // [128,128] row-major (D_OUT, D_IN)
    const float* __restrict__ b_l,      // [128]
    const float* __restrict__ W_r,      // [128,128]
    const float* __restrict__ W_head,   // [128]  (shape (1,128))
    const float* __restrict__ b_head,   // [1]
    float* __restrict__ out)            // [n_nodes]
{
    // +4 float row pad: rows start 16B-aligned and the 16-lane column access
    // of the A-fragment spreads across all 64 LDS banks (no conflicts).
    __shared__ alignas(16) float lds_mean[16][D + 4];
    __shared__ alignas(16) float lds_x[16][D + 4];
    __shared__ float lds_part[8][16];

    const int tid   = threadIdx.x;
    const int node0 = blockIdx.x * 16;

    // ---- stage mean (= sums / max(cnt,1)) and x_dst tiles into LDS --------
    for (int i = tid; i < 16 * (D / 4); i += 256) {
        int row  = i >> 5;          // 0..15
        int kq   = (i & 31) << 2;   // 0,4,...,124
        int node = node0 + row;
        float inv = 1.0f / fmaxf(cnt[node], 1.0f);
        const float4 s4 = *(const float4*)(sums  + (size_t)node * D + kq);
        const float4 x4 = *(const float4*)(x_dst + (size_t)node * D + kq);
        *(float4*)(&lds_mean[row][kq]) =
            make_float4(s4.x * inv, s4.y * inv, s4.z * inv, s4.w * inv);
        *(float4*)(&lds_x[row][kq]) = x4;
    }
    __syncthreads();

    const int wave = tid >> 5;
    const int lane = tid & 31;
    const int half = lane >> 4;   // selects K pair {0,1} vs {2,3}
    const int lrow = lane & 15;   // M for A-frag, N for B-frag / C column
    const int n    = wave * 16 + lrow;   // this lane's output feature column

    const float* __restrict__ Wl_p = W_l + (size_t)n * D + 2 * half;
    const float* __restrict__ Wr_p = W_r + (size_t)n * D + 2 * half;
    const float* __restrict__ m_p  = &lds_mean[lrow][2 * half];
    const float* __restrict__ x_p  = &lds_x[lrow][2 * half];

    v8f acc = {};   // 16x16 f32 accumulator (8 VGPRs)

#pragma unroll 4
    for (int k0 = 0; k0 < D; k0 += 4) {
        v2f am = { m_p[k0], m_p[k0 + 1] };                 // mean A-frag (LDS)
        v2f bl = { Wl_p[k0], Wl_p[k0 + 1] };               // W_l.T B-frag
        acc = __builtin_amdgcn_wmma_f32_16x16x4_f32(
            false, am, false, bl, (short)0, acc, false, false);

        v2f ax = { x_p[k0], x_p[k0 + 1] };                 // x_dst A-frag (LDS)
        v2f br = { Wr_p[k0], Wr_p[k0 + 1] };               // W_r.T B-frag
        acc = __builtin_amdgcn_wmma_f32_16x16x4_f32(
            false, ax, false, br, (short)0, acc, false, false);
    }

    // ---- bias + ReLU + head partial dot (this wave covers 16 of 128 feats)
    const float bias = b_l[n];
    const float wh   = W_head[n];
#pragma unroll
    for (int r = 0; r < 8; ++r) {
        float h = acc[r] + bias;
        h = fmaxf(h, 0.0f);                  // ReLU on hidden feature
        float p = h * wh;                    // head contribution of column n
        // sum across the 16 lanes of this half-wave (over N)
        p += __shfl_xor(p, 1, 32);
        p += __shfl_xor(p, 2, 32);
        p += __shfl_xor(p, 4, 32);
        p += __shfl_xor(p, 8, 32);
        if (lrow == 0) lds_part[wave][r + 8 * half] = p;   // row m = r + 8*half
    }
    __syncthreads();

    // ---- combine the 8 wave partials, leaky-ReLU, store scalar ------------
    if (tid < 16) {
        float dot = 0.0f;
#pragma unroll
        for (int w = 0; w < 8; ++w) dot += lds_part[w][tid];
        dot += b_head[0];
        out[node0 + tid] = (dot > 0.0f) ? dot : 0.001f * dot;
    }
}

// ---------------------------------------------------------------------------
extern "C" void kernel_launch(void* const* d_in, const int* in_sizes, int n_in,
                              void* d_out, int out_size, void* d_ws, size_t ws_size,
                              hipStream_t stream)
{
    const float* x_sites = (const float*)d_in[0];
    const float* x_wells = (const float*)d_in[1];
    const float* W_l_s2g = (const float*)d_in[2];
    const float* b_l_s2g = (const float*)d_in[3];
    const float* W_r_s2g = (const float*)d_in[4];
    const float* W_l_g2s = (const float*)d_in[5];
    const float* b_l_g2s = (const float*)d_in[6];
    const float* W_r_g2s = (const float*)d_in[7];
    const float* W_gw    = (const float*)d_in[8];
    const float* b_gw    = (const float*)d_in[9];
    const float* W_sites = (const float*)d_in[10];
    const float* b_sites = (const float*)d_in[11];
    const int* e_s2g_src = (const int*)d_in[12];
    const int* e_s2g_dst = (const int*)d_in[13];
    const int* e_g2s_src = (const int*)d_in[14];
    const int* e_g2s_dst = (const int*)d_in[15];

    float* out = (float*)d_out;   // [0..N_WELLS) = out_gw, then out_sites

    // workspace layout (floats)
    float* ws      = (float*)d_ws;
    float* sums_w  = ws;                                   // N_WELLS*D
    float* cnt_w   = sums_w + (size_t)N_WELLS * D;         // N_WELLS
    float* sums_s  = cnt_w + N_WELLS;                      // N_SITES*D
    float* cnt_s   = sums_s + (size_t)N_SITES * D;         // N_SITES
    size_t ws_floats = (size_t)N_WELLS * D + N_WELLS + (size_t)N_SITES * D + N_SITES;

    // zero the accumulation scratch (graph-capture safe)
    hipMemsetAsync(d_ws, 0, ws_floats * sizeof(float), stream);

    // Phase 1: scatter-mean accumulation (1 wave per edge)
    {
        long long threads = (long long)N_EDGES * 32;
        int blocks = (int)((threads + 255) / 256);
        scatter_edges_kernel<<<blocks, 256, 0, stream>>>(
            x_sites, e_s2g_src, e_s2g_dst, sums_w, cnt_w, N_EDGES);
        scatter_edges_kernel<<<blocks, 256, 0, stream>>>(
            x_wells, e_g2s_src, e_g2s_dst, sums_s, cnt_s, N_EDGES);
    }

    // Phase 2: fused WMMA node update + head
    node_update_wmma_kernel<<<N_WELLS / 16, 256, 0, stream>>>(
        sums_w, cnt_w, x_wells, W_l_s2g, b_l_s2g, W_r_s2g,
        W_gw, b_gw, out);
    node_update_wmma_kernel<<<N_SITES / 16, 256, 0, stream>>>(
        sums_s, cnt_s, x_sites, W_l_g2s, b_l_g2s, W_r_g2s,
        W_sites, b_sites, out + N_WELLS);
}